// Classifier_heterogeneous_irt_76209899701047
// MI455X (gfx1250) — compile-verified
//
#include <hip/hip_runtime.h>
#include <hip/hip_bf16.h>

#define N_STUDENT 100000
#define N_ITEM    20000
#define N_EDGES   1000000
#define IN_CH     128
#define EDGE_DIM  32
#define CAT_CH    (IN_CH + EDGE_DIM)   // 160
#define DEC_CH    64

typedef float v2f __attribute__((ext_vector_type(2)));
typedef float v8f __attribute__((ext_vector_type(8)));

__device__ __forceinline__ float elu_f(float x) {
    return x > 0.0f ? x : (__expf(x) - 1.0f);
}
__device__ __forceinline__ float softplus_f(float x) {
    // stable: max(x,0) + log1p(exp(-|x|))
    return fmaxf(x, 0.0f) + __logf(1.0f + __expf(-fabsf(x)));
}

// One wave (32 lanes) computes a 16-edge tile:
//   X[16x64] = elu([feat_s | edge_feat] @ W1 + b1)   (K = 160)
//   Y[16x64] = softplus(feat_i @ W2 + b2)            (K = 128)
//   out[16]  = rowsum(X * Y) + offset[dst]
// using V_WMMA_F32_16X16X4_F32 (fp32-exact matrix path on CDNA5).
__global__ __launch_bounds__(256)
void irt_decoder_kernel(const float* __restrict__ xs,   // [N_STUDENT, IN_CH]
                        const float* __restrict__ xi,   // [N_ITEM, IN_CH]
                        const int*   __restrict__ eli,  // [2, N_EDGES]
                        const float* __restrict__ ef,   // [N_EDGES, EDGE_DIM]
                        const float* __restrict__ off,  // [N_ITEM, 1]
                        const float* __restrict__ W1,   // [CAT_CH, DEC_CH]
                        const float* __restrict__ b1,   // [DEC_CH]
                        const float* __restrict__ W2,   // [IN_CH, DEC_CH]
                        const float* __restrict__ b2,   // [DEC_CH]
                        float*       __restrict__ out)  // [N_EDGES, 1]
{
    const int lane = threadIdx.x & 31;
    const int wave = threadIdx.x >> 5;
    const int tile = blockIdx.x * 8 + wave;
    const int ntiles = N_EDGES / 16;          // 62500, exact
    if (tile >= ntiles) return;               // wave-uniform: EXEC stays all-ones

    const int e0   = tile * 16;
    const int row  = lane & 15;               // A-matrix row (edge within tile), also B/C column-in-tile
    const int half = lane >> 4;               // 0: K pair {0,1}; 1: K pair {2,3}
    const int e    = e0 + row;
    const int src  = eli[e];                  // student index for this lane's row
    const int dst  = eli[N_EDGES + e];        // item index for this lane's row

    // ---------------- Y = x_item[dst] @ W2, K = 128 ----------------
    v8f accy[4] = {};
    {
        const float* arow = xi + (long)dst * IN_CH;
        #pragma unroll 2
        for (int k0 = 0; k0 < IN_CH; k0 += 4) {
            const int ka = k0 + 2 * half;            // this lane's K for A (and B reg 0)
            v2f a = *(const v2f*)(arow + ka);        // A: (row, ka), (row, ka+1)
            #pragma unroll
            for (int n = 0; n < 4; ++n) {
                const int col = n * 16 + row;        // B column
                v2f b;
                b.x = W2[ka * DEC_CH + col];         // B: (ka,   col)
                b.y = W2[(ka + 1) * DEC_CH + col];   // B: (ka+1, col)
                accy[n] = __builtin_amdgcn_wmma_f32_16x16x4_f32(
                    false, a, false, b, (short)0, accy[n], false, false);
            }
        }
    }
    // bias + softplus, keep in registers
    v8f spy[4];
    #pragma unroll
    for (int n = 0; n < 4; ++n) {
        const float bv = b2[n * 16 + row];
        #pragma unroll
        for (int r = 0; r < 8; ++r) spy[n][r] = softplus_f(accy[n][r] + bv);
    }

    // ------------- X = [feat_s | edge_feat] @ W1, K = 160 -------------
    v8f accx[4] = {};
    {
        const float* rowS = xs + (long)src * IN_CH;
        const float* rowE = ef + ((long)e * EDGE_DIM - IN_CH); // +col gives ef[e][col-128]
        #pragma unroll 2
        for (int k0 = 0; k0 < CAT_CH; k0 += 4) {
            const float* rp = (k0 < IN_CH) ? rowS : rowE;      // uniform per k-step
            const int ka = k0 + 2 * half;
            v2f a = *(const v2f*)(rp + ka);
            #pragma unroll
            for (int n = 0; n < 4; ++n) {
                const int col = n * 16 + row;
                v2f b;
                b.x = W1[ka * DEC_CH + col];
                b.y = W1[(ka + 1) * DEC_CH + col];
                accx[n] = __builtin_amdgcn_wmma_f32_16x16x4_f32(
                    false, a, false, b, (short)0, accx[n], false, false);
            }
        }
    }

    // ---- per-lane partial row sums: part[r] = sum over this lane's 4 columns ----
    // C/D layout: lanes 0-15 hold rows r (reg r), lanes 16-31 hold rows r+8.
    float part[8];
    #pragma unroll
    for (int r = 0; r < 8; ++r) part[r] = 0.0f;
    #pragma unroll
    for (int n = 0; n < 4; ++n) {
        const float bv = b1[n * 16 + row];
        #pragma unroll
        for (int r = 0; r < 8; ++r)
            part[r] += elu_f(accx[n][r] + bv) * spy[n][r];
    }

    // butterfly across the 16 lanes of each half (masks < 16 keep halves separate)
    #pragma unroll
    for (int m = 8; m >= 1; m >>= 1) {
        #pragma unroll
        for (int r = 0; r < 8; ++r) part[r] += __shfl_xor(part[r], m, 32);
    }

    // lanes with (lane%16) < 8 each emit one row:
    //   lanes 0..7  -> rows 0..7, lanes 16..23 -> rows 8..15
    if (row < 8) {
        const int s = lane & 7;
        float v = part[0];
        #pragma unroll
        for (int r = 1; r < 8; ++r) v = (s == r) ? part[r] : v;
        const int m  = (half << 3) | s;          // row in tile
        const int em = e0 + m;                   // global edge id
        const int d  = eli[N_EDGES + em];        // item index for that row
        out[em] = v + off[d];
    }
}

extern "C" void kernel_launch(void* const* d_in, const int* in_sizes, int n_in,
                              void* d_out, int out_size, void* d_ws, size_t ws_size,
                              hipStream_t stream) {
    const float* xs  = (const float*)d_in[0];  // x_student
    const float* xi  = (const float*)d_in[1];  // x_item
    const int*   eli = (const int*)  d_in[2];  // edge_label_index [2, E]
    const float* ef  = (const float*)d_in[3];  // edge_feat
    const float* off = (const float*)d_in[4];  // offset
    const float* W1  = (const float*)d_in[5];
    const float* b1  = (const float*)d_in[6];
    const float* W2  = (const float*)d_in[7];
    const float* b2  = (const float*)d_in[8];
    float* out = (float*)d_out;

    const int ntiles = N_EDGES / 16;           // 62500
    const int blocks = (ntiles + 7) / 8;       // 8 waves (tiles) per 256-thread block
    irt_decoder_kernel<<<blocks, 256, 0, stream>>>(xs, xi, eli, ef, off, W1, b1, W2, b2, out);
}